// VecEdgesRead_22651657519348
// MI455X (gfx1250) — compile-verified
//
#include <hip/hip_runtime.h>

// ---------------------------------------------------------------------------
// VecEdgesRead: out[b,e,o,k] = (Wsrc@x[b,src[e]] + Wdst@x[b,dst[e]])[o,k]/sqrt(2)
// B=2, DIN=DOUT=32, k=3.  Strategy: precompute per-node transforms with
// V_WMMA_F32_16X16X4_F32 (fp32 tensor path), then a streaming gather+add with
// non-temporal b128 stores.
// ---------------------------------------------------------------------------

typedef __attribute__((ext_vector_type(2))) float v2f;
typedef __attribute__((ext_vector_type(4))) float vf4;
typedef __attribute__((ext_vector_type(8))) float v8f;

#define INV_SQRT_2 0.70710678118654752440f
#define ROW96 96   // DIN*3 floats per node (also DOUT*3 per transformed node)

// ---------------------------------------------------------------------------
// Pass 1: node transform.  Y[b][n][o*3+k] = sum_i W[o][i] * x[b][n][i*3+k]
// One wave computes a 16(o) x 16(columns of flattened (n,k)) fp32 tile,
// looping K=32 in 8 WMMA 16x16x4 steps.  8 waves/block cover
// {Wsrc,Wdst} x {b} x {o-tile}.  blockIdx.x = column tile (NK/16 tiles).
// ---------------------------------------------------------------------------
__global__ __launch_bounds__(256) void node_transform_kernel(
    const float* __restrict__ x,
    const float* __restrict__ Wsrc,
    const float* __restrict__ Wdst,
    float* __restrict__ ysrc,
    float* __restrict__ ydst,
    int NK /* = N*3 */) {
  const int lane  = threadIdx.x & 31;
  const int wave  = threadIdx.x >> 5;
  const int wsel  = wave & 1;         // 0 -> Wsrc/ysrc, 1 -> Wdst/ydst
  const int b     = (wave >> 1) & 1;  // batch
  const int ot    = (wave >> 2) & 1;  // o-tile (0..1), 16 rows each
  const int m     = lane & 15;        // row (A/D) or column (B/D) index
  const int khalf = (lane >> 4) << 1; // 0 for lanes 0-15, 2 for lanes 16-31

  const float* __restrict__ W = wsel ? Wdst : Wsrc;
  float* __restrict__ Y       = wsel ? ydst : ysrc;

  // Column of the flattened (n,k) axis handled by this lane (B/D operand).
  int c = blockIdx.x * 16 + m;
  if (c >= NK) c = NK - 1;  // clamped lanes duplicate the last column; the
                            // duplicated D values are identical -> benign.
  const int n = c / 3;
  const int k = c - n * 3;

  const long long bstride = (long long)NK * 32;  // N*96 floats per batch
  const float* __restrict__ xcol = x + (long long)b * bstride + (long long)n * ROW96 + k;
  const float* __restrict__ Wrow = W + (ot * 16 + m) * 32;

  v8f acc = {0.f, 0.f, 0.f, 0.f, 0.f, 0.f, 0.f, 0.f};

#pragma unroll
  for (int kk = 0; kk < 8; ++kk) {
    const int i0 = kk * 4 + khalf;  // K index for VGPR0 of A and B operands
    v2f a, bm;
    // A (16x4 f32): lanes 0-15 hold M=lane {K=i0,i0+1}; lanes 16-31 same M
    // with {K=i0,i0+1} where i0 has khalf=2.  (ISA 7.12.2 layout.)
    a.x = Wrow[i0];
    a.y = Wrow[i0 + 1];
    // B (4x16 f32): mirrored layout, N=lane&15 striped across lanes.
    bm.x = xcol[i0 * 3];
    bm.y = xcol[i0 * 3 + 3];
    acc = __builtin_amdgcn_wmma_f32_16x16x4_f32(
        /*neg_a=*/false, a, /*neg_b=*/false, bm,
        /*c_mod=*/(short)0, acc, /*reuse_a=*/false, /*reuse_b=*/false);
  }

  // D layout (32-bit 16x16): VGPR v -> M = v + (lane<16 ? 0 : 8), N = lane&15.
  const long long ybase = (long long)b * bstride + (long long)n * ROW96 + k;
  const int o_base = ot * 16 + (khalf << 2);  // +8 for hi half-lanes
#pragma unroll
  for (int v = 0; v < 8; ++v) {
    Y[ybase + (long long)(o_base + v) * 3] = acc[v];
  }
}

// ---------------------------------------------------------------------------
// Pass 2: streaming gather + add.  Each block handles 10 edges for one batch;
// 24 threads per edge move 96 floats as float4.  Output writes are fully
// coalesced and use non-temporal b128 stores so the 614 MB output stream does
// not evict the 76.8 MB gather tables from L2.
// ---------------------------------------------------------------------------
__global__ __launch_bounds__(256) void edge_gather_kernel(
    const vf4* __restrict__ ysrc,
    const vf4* __restrict__ ydst,
    const int* __restrict__ src,
    const int* __restrict__ dst,
    vf4* __restrict__ out,
    int N, int E) {
  const int tid = threadIdx.x;
  if (tid >= 240) return;                       // 10 edges * 24 quads
  const int e = blockIdx.x * 10 + tid / 24;     // small-const div -> mul/shift
  if (e >= E) return;
  const int q = tid % 24;
  const int b = blockIdx.y;

  const long long nbase = (long long)b * N;
  const long long sOff  = (nbase + src[e]) * 24 + q;
  const long long dOff  = (nbase + dst[e]) * 24 + q;

  const vf4 a = ysrc[sOff];
  const vf4 c = ydst[dOff];
  const vf4 r = (a + c) * INV_SQRT_2;

  const long long be = (long long)b * E + e;
  __builtin_nontemporal_store(r, out + be * 24 + q);
}

// ---------------------------------------------------------------------------
// Fallback (only if workspace is too small): direct per-edge compute.
// One thread per (b,e,o) produces 3 outputs.
// ---------------------------------------------------------------------------
__global__ __launch_bounds__(256) void direct_edge_kernel(
    const float* __restrict__ x,
    const int* __restrict__ src,
    const int* __restrict__ dst,
    const float* __restrict__ Wsrc,
    const float* __restrict__ Wdst,
    float* __restrict__ out,
    int N, int E) {
  const long long idx = (long long)blockIdx.x * 256 + threadIdx.x;
  const long long total = 2LL * E * 32;
  if (idx >= total) return;
  const int o = (int)(idx & 31);
  const long long be = idx >> 5;
  const int e = (int)(be % E);
  const int b = (int)(be / E);

  const float* __restrict__ xs = x + ((long long)b * N + src[e]) * ROW96;
  const float* __restrict__ xd = x + ((long long)b * N + dst[e]) * ROW96;
  const float* __restrict__ ws = Wsrc + o * 32;
  const float* __restrict__ wd = Wdst + o * 32;

  float a0 = 0.f, a1 = 0.f, a2 = 0.f;
#pragma unroll 8
  for (int i = 0; i < 32; ++i) {
    const float s = ws[i], d = wd[i];
    a0 += s * xs[i * 3 + 0] + d * xd[i * 3 + 0];
    a1 += s * xs[i * 3 + 1] + d * xd[i * 3 + 1];
    a2 += s * xs[i * 3 + 2] + d * xd[i * 3 + 2];
  }
  float* op = out + (be * 32 + o) * 3;
  op[0] = a0 * INV_SQRT_2;
  op[1] = a1 * INV_SQRT_2;
  op[2] = a2 * INV_SQRT_2;
}

// ---------------------------------------------------------------------------
extern "C" void kernel_launch(void* const* d_in, const int* in_sizes, int n_in,
                              void* d_out, int out_size, void* d_ws, size_t ws_size,
                              hipStream_t stream) {
  const float* x    = (const float*)d_in[0];
  const int*   src  = (const int*)d_in[1];
  const int*   dst  = (const int*)d_in[2];
  const float* Wsrc = (const float*)d_in[3];
  const float* Wdst = (const float*)d_in[4];
  float*       out  = (float*)d_out;

  const long long BN = (long long)in_sizes[0] / ROW96;  // B*N
  const int B  = 2;                                      // reference hardcodes B=2
  const int N  = (int)(BN / B);
  const int E  = in_sizes[1];
  const int NK = N * 3;

  const size_t table_floats = (size_t)B * N * ROW96;       // per table
  const size_t need = 2 * table_floats * sizeof(float);    // ysrc + ydst

  if (ws_size >= need) {
    float* ysrc = (float*)d_ws;
    float* ydst = ysrc + table_floats;

    // Pass 1: WMMA node transform (NK/16 column tiles, 8 waves per block).
    const int ctiles = (NK + 15) / 16;
    node_transform_kernel<<<ctiles, 256, 0, stream>>>(x, Wsrc, Wdst, ysrc, ydst, NK);

    // Pass 2: streaming gather+add, 10 edges per block, one batch per grid.y.
    dim3 grid((E + 9) / 10, B);
    edge_gather_kernel<<<grid, 256, 0, stream>>>(
        (const vf4*)ysrc, (const vf4*)ydst, src, dst, (vf4*)out, N, E);
  } else {
    const long long total = 2LL * E * 32;
    const int blocks = (int)((total + 255) / 256);
    direct_edge_kernel<<<blocks, 256, 0, stream>>>(x, src, dst, Wsrc, Wdst, out, N, E);
  }
}